// GCNEncoder_26723286515821
// MI455X (gfx1250) — compile-verified
//
#include <hip/hip_runtime.h>

#define HDIM 128
#define BN_EPS 1e-5f

typedef __attribute__((ext_vector_type(2))) float v2f;
typedef __attribute__((ext_vector_type(8))) float v8f;

// ---------------------------------------------------------------------------
// Degree / normalization
// ---------------------------------------------------------------------------
__global__ void k_init(float* __restrict__ deg, float* __restrict__ stats, int n) {
    int i = blockIdx.x * blockDim.x + threadIdx.x;
    if (i < n) deg[i] = 1.0f;          // self-loop contributes 1 to every node
    if (i < 256) stats[i] = 0.0f;      // zero BN accumulators for layer 1
}

__global__ void k_deg(const int* __restrict__ dst, float* __restrict__ deg, int ne) {
    int e = blockIdx.x * blockDim.x + threadIdx.x;
    if (e < ne) unsafeAtomicAdd(&deg[dst[e]], 1.0f);   // global_atomic_add_f32
}

__global__ void k_dinv(float* __restrict__ deg, int n) {
    int i = blockIdx.x * blockDim.x + threadIdx.x;
    if (i < n) deg[i] = rsqrtf(fmaxf(deg[i], 1.0f));
}

// ---------------------------------------------------------------------------
// Dense transform H = X @ W via V_WMMA_F32_16X16X4_F32 (fp32, wave32)
// One wave computes one 16x16 output tile; K = 128 in steps of 4.
// A (16x4 f32): lanes 0-15 hold M=lane, K={0,1}; lanes 16-31 hold M=lane-16, K={2,3}
// B (4x16 f32): lanes 0-15 hold N=lane, K={0,1}; lanes 16-31 hold N=lane-16, K={2,3}
// C/D (16x16 f32): VGPR r: lanes 0-15 -> (M=r, N=lane); lanes 16-31 -> (M=8+r, N=lane-16)
// ---------------------------------------------------------------------------
__global__ void __launch_bounds__(256) k_gemm(const float* __restrict__ X,
                                              const float* __restrict__ W,
                                              float* __restrict__ H, int nrows) {
    const int lane = threadIdx.x & 31;
    const int wave = threadIdx.x >> 5;
    const int tile = blockIdx.x * 8 + wave;
    const int m0 = (tile >> 3) * 16;       // HDIM/16 == 8 column tiles
    const int n0 = (tile & 7) * 16;
    if (m0 >= nrows) return;               // wave-uniform: EXEC stays all-1s
    const int lm = lane & 15;
    const int kk = (lane >> 4) * 2;        // 0 for lanes 0-15, 2 for lanes 16-31

    const float* xp = X + (size_t)(m0 + lm) * HDIM + kk;
    const float* wp = W + (size_t)kk * HDIM + n0 + lm;

    v8f c = {};
#pragma unroll 4
    for (int k = 0; k < HDIM; k += 4) {
        float2 av = *(const float2*)(xp + k);      // K = k+kk, k+kk+1 (8B aligned)
        v2f a, b;
        a.x = av.x;
        a.y = av.y;
        b.x = wp[(size_t)k * HDIM];                // K = k+kk
        b.y = wp[(size_t)k * HDIM + HDIM];         // K = k+kk+1
        c = __builtin_amdgcn_wmma_f32_16x16x4_f32(false, a, false, b,
                                                  (short)0, c, false, false);
    }

    const int mb = m0 + (lane >> 4) * 8;
#pragma unroll
    for (int r = 0; r < 8; ++r)
        H[(size_t)(mb + r) * HDIM + n0 + lm] = c[r];
}

// ---------------------------------------------------------------------------
// Aggregation: out[i] = b + dinv[i]^2 * h[i]  (self-loop term + bias init)
// ---------------------------------------------------------------------------
__global__ void k_agg_init(const float* __restrict__ H, const float* __restrict__ bias,
                           const float* __restrict__ dinv, float* __restrict__ out,
                           int total) {
    int i = blockIdx.x * blockDim.x + threadIdx.x;
    if (i < total) {
        int row = i >> 7;
        int col = i & (HDIM - 1);
        float di = dinv[row];
        out[i] = bias[col] + H[i] * di * di;
    }
}

// One wave per edge: gather 128 floats of h[src] (float4/lane), scale by
// dinv[src]*dinv[dst], scatter-add into out[dst] with f32 atomics (L2-resident).
__global__ void __launch_bounds__(256) k_agg_edges(const float* __restrict__ H,
                                                   const int* __restrict__ src,
                                                   const int* __restrict__ dst,
                                                   const float* __restrict__ dinv,
                                                   float* __restrict__ out, int ne) {
    const int lane = threadIdx.x & 31;
    const int wave = threadIdx.x >> 5;
    const int e = blockIdx.x * 8 + wave;
    if (e >= ne) return;
    const int s = src[e];
    const int d = dst[e];
    const float norm = dinv[s] * dinv[d];
    const float4 v = ((const float4*)(H + (size_t)s * HDIM))[lane];
    float* o = out + (size_t)d * HDIM + lane * 4;
    unsafeAtomicAdd(o + 0, v.x * norm);
    unsafeAtomicAdd(o + 1, v.y * norm);
    unsafeAtomicAdd(o + 2, v.z * norm);
    unsafeAtomicAdd(o + 3, v.w * norm);
}

// ---------------------------------------------------------------------------
// BatchNorm: column sum / sumsq, then fused normalize + ReLU
// ---------------------------------------------------------------------------
__global__ void k_zero_stats(float* __restrict__ stats) {
    stats[threadIdx.x] = 0.0f;   // 256 threads
}

__global__ void k_bn_stats(const float* __restrict__ X, float* __restrict__ stats,
                           int n) {
    const int col = threadIdx.x;           // blockDim.x == 128
    float s = 0.0f, sq = 0.0f;
    for (int r = blockIdx.x; r < n; r += gridDim.x) {
        float v = X[(size_t)r * HDIM + col];
        s += v;
        sq += v * v;
    }
    unsafeAtomicAdd(&stats[col], s);
    unsafeAtomicAdd(&stats[HDIM + col], sq);
}

__global__ void k_bn_relu(const float* __restrict__ X, const float* __restrict__ stats,
                          const float* __restrict__ gamma, const float* __restrict__ beta,
                          float* __restrict__ out, int n, int total) {
    int i = blockIdx.x * blockDim.x + threadIdx.x;
    if (i < total) {
        int col = i & (HDIM - 1);
        float inv_n = 1.0f / (float)n;
        float mu = stats[col] * inv_n;
        float var = stats[HDIM + col] * inv_n - mu * mu;
        float v = (X[i] - mu) * rsqrtf(var + BN_EPS) * gamma[col] + beta[col];
        out[i] = fmaxf(v, 0.0f);
    }
}

// ---------------------------------------------------------------------------
// Launch: inputs = node_feat, edge_index, W1, b1, W2, b2, g1, beta1, g2, beta2
// ---------------------------------------------------------------------------
extern "C" void kernel_launch(void* const* d_in, const int* in_sizes, int n_in,
                              void* d_out, int out_size, void* d_ws, size_t ws_size,
                              hipStream_t stream) {
    const float* x0  = (const float*)d_in[0];
    const int*   ei  = (const int*)d_in[1];
    const float* W1  = (const float*)d_in[2];
    const float* b1  = (const float*)d_in[3];
    const float* W2  = (const float*)d_in[4];
    const float* b2  = (const float*)d_in[5];
    const float* g1  = (const float*)d_in[6];
    const float* be1 = (const float*)d_in[7];
    const float* g2  = (const float*)d_in[8];
    const float* be2 = (const float*)d_in[9];

    const int n  = in_sizes[0] / HDIM;   // 50000 nodes
    const int ne = in_sizes[1] / 2;      // 800000 edges
    const int* src = ei;                 // edge_index row 0
    const int* dst = ei + ne;            // edge_index row 1
    const int total = n * HDIM;

    float* ws    = (float*)d_ws;
    float* dinv  = ws;                               // [n]
    float* stats = ws + n;                           // [256]
    float* hbuf  = stats + 256;                      // [n*HDIM]
    float* abuf  = hbuf + (size_t)n * HDIM;          // [n*HDIM]
    float* out   = (float*)d_out;                    // [n*HDIM]

    const dim3 B(256);
    const int rtiles  = (n + 15) / 16;
    const int tiles   = rtiles * (HDIM / 16);
    const int gblocks = (tiles + 7) / 8;
    const int eblocks = (ne + 7) / 8;
    const int tblocks = (total + 255) / 256;

    // Degree -> dinv (shared by both layers); zero BN stats for layer 1
    k_init <<<(n + 255) / 256, B, 0, stream>>>(dinv, stats, n);
    k_deg  <<<(ne + 255) / 256, B, 0, stream>>>(dst, dinv, ne);
    k_dinv <<<(n + 255) / 256, B, 0, stream>>>(dinv, n);

    // ---- Layer 1 ----
    k_gemm      <<<gblocks, B, 0, stream>>>(x0, W1, hbuf, n);
    k_agg_init  <<<tblocks, B, 0, stream>>>(hbuf, b1, dinv, abuf, total);
    k_agg_edges <<<eblocks, B, 0, stream>>>(hbuf, src, dst, dinv, abuf, ne);
    k_bn_stats  <<<256, 128, 0, stream>>>(abuf, stats, n);
    k_bn_relu   <<<tblocks, B, 0, stream>>>(abuf, stats, g1, be1, hbuf, n, total);

    // ---- Layer 2 ----
    k_gemm       <<<gblocks, B, 0, stream>>>(hbuf, W2, abuf, n);
    k_agg_init   <<<tblocks, B, 0, stream>>>(abuf, b2, dinv, out, total);
    k_agg_edges  <<<eblocks, B, 0, stream>>>(abuf, src, dst, dinv, out, ne);
    k_zero_stats <<<1, 256, 0, stream>>>(stats);
    k_bn_stats   <<<256, 128, 0, stream>>>(out, stats, n);
    k_bn_relu    <<<tblocks, B, 0, stream>>>(out, stats, g2, be2, out, n, total);
}